// IntegratePointWiseNeurips_901943132677
// MI455X (gfx1250) — compile-verified
//
#include <hip/hip_runtime.h>

#define GRIDN     128
#define NUM_STEPS 7
#define NPTS      200000
#define NBATCH    2
#define NVOX      (GRIDN * GRIDN * GRIDN)   // 2^21 voxels per batch
#define VSHIFT    21                        // log2(NVOX)

// ---------------------------------------------------------------------------
// Pass 1a: channel-interleave + pad  flow[b][c][v] -> vol4[b][v] (float4)
// Exact-sized grid, pure 32-bit indexing. flow is read exactly once ->
// non-temporal loads keep it from competing with vol4 for L2 residency.
// Stores are consecutive 16B per lane -> coalesced global_store_b128.
// ---------------------------------------------------------------------------
__global__ __launch_bounds__(256)
void pack4_kernel(const float* __restrict__ flow, float4* __restrict__ vol) {
    const int t = blockIdx.x * blockDim.x + threadIdx.x;   // [0, NBATCH*NVOX)
    const int b = t >> VSHIFT;
    const int v = t & (NVOX - 1);
    const float* __restrict__ src = flow + ((long long)b * 3 << VSHIFT) + v;
    float4 o;
    o.x = __builtin_nontemporal_load(src);
    o.y = __builtin_nontemporal_load(src + NVOX);
    o.z = __builtin_nontemporal_load(src + 2 * NVOX);
    o.w = 0.0f;
    vol[t] = o;
}

// ---------------------------------------------------------------------------
// Pass 1b: 12-byte interleave (fallback if ws < 64MB but >= 48MB)
// ---------------------------------------------------------------------------
__global__ __launch_bounds__(256)
void pack3_kernel(const float* __restrict__ flow, float* __restrict__ vol) {
    const int t = blockIdx.x * blockDim.x + threadIdx.x;
    const int b = t >> VSHIFT;
    const int v = t & (NVOX - 1);
    const float* __restrict__ src = flow + ((long long)b * 3 << VSHIFT) + v;
    float* __restrict__ dst = vol + (long long)t * 3;
    dst[0] = __builtin_nontemporal_load(src);
    dst[1] = __builtin_nontemporal_load(src + NVOX);
    dst[2] = __builtin_nontemporal_load(src + 2 * NVOX);
}

// ---------------------------------------------------------------------------
// Pass 2: scaling-and-squaring integration, one thread per point.
// LAYOUT 2: float4-padded volume  -> 8 single b128 gathers per step (best)
// LAYOUT 1: 12B interleaved       -> 8 b64+b32 gathers per step
// LAYOUT 0: planar original       -> 24 b32 gathers per step (no ws needed)
// ---------------------------------------------------------------------------
template <int LAYOUT>
__global__ __launch_bounds__(256)
void integrate_kernel(const float*  __restrict__ fieldF,
                      const float4* __restrict__ fieldV,
                      const float*  __restrict__ x,
                      float* __restrict__ out) {
    const int t = blockIdx.x * blockDim.x + threadIdx.x;
    if (t >= NBATCH * NPTS) return;
    const int b = t / NPTS;

    const float4* __restrict__ basev =
        (LAYOUT == 2) ? fieldV + ((long long)b << VSHIFT) : nullptr;
    const float* __restrict__ basef =
        (LAYOUT == 2) ? nullptr : fieldF + ((long long)b * 3 << VSHIFT);

    // coords: (d, h, w) per reference; channel c of flow displaces coord c
    float pd = x[3 * t + 0];
    float ph = x[3 * t + 1];
    float pw = x[3 * t + 2];

    const float scale = 1.0f / 128.0f;     // 2^-NUM_STEPS folded into output
    const float lim   = (float)(GRIDN - 1);

    #pragma unroll 1
    for (int s = 0; s < NUM_STEPS; ++s) {
        // border padding == clamp coords (align_corners=True identity mapping)
        const float d = fminf(fmaxf(pd, 0.0f), lim);
        const float h = fminf(fmaxf(ph, 0.0f), lim);
        const float w = fminf(fmaxf(pw, 0.0f), lim);

        const int d0 = (int)d;             // floor: value >= 0
        const int h0 = (int)h;
        const int w0 = (int)w;
        const int d1 = min(d0 + 1, GRIDN - 1);
        const int h1 = min(h0 + 1, GRIDN - 1);
        const int w1 = min(w0 + 1, GRIDN - 1);

        const float fd = d - (float)d0;
        const float fh = h - (float)h0;
        const float fw = w - (float)w0;

        // 8 corner voxel indices (cheap add-form: i000 + {0,dw,dh,dd} combos)
        const int i000 = (d0 * GRIDN + h0) * GRIDN + w0;
        const int dw = w1 - w0;                 // 0 or 1
        const int dh = (h1 - h0) * GRIDN;       // 0 or 128
        const int dd = (d1 - d0) * GRIDN * GRIDN;

        float c[8][3];
        const int idx[8] = { i000,           i000 + dw,
                             i000 + dh,      i000 + dh + dw,
                             i000 + dd,      i000 + dd + dw,
                             i000 + dd + dh, i000 + dd + dh + dw };

        if (LAYOUT == 2) {
            #pragma unroll
            for (int k = 0; k < 8; ++k) {
                const float4 q = basev[idx[k]];   // one global_load_b128
                c[k][0] = q.x; c[k][1] = q.y; c[k][2] = q.z;
            }
        } else if (LAYOUT == 1) {
            #pragma unroll
            for (int k = 0; k < 8; ++k) {
                const float* p = basef + (long long)idx[k] * 3;
                c[k][0] = p[0]; c[k][1] = p[1]; c[k][2] = p[2];
            }
        } else {
            #pragma unroll
            for (int ch = 0; ch < 3; ++ch) {
                const float* __restrict__ plane = basef + ((long long)ch << VSHIFT);
                #pragma unroll
                for (int k = 0; k < 8; ++k) c[k][ch] = plane[idx[k]];
            }
        }

        // trilinear blend per channel: a + t*(b-a) form (sub + fma)
        float r[3];
        #pragma unroll
        for (int ch = 0; ch < 3; ++ch) {
            const float c00 = fmaf(fw, c[1][ch] - c[0][ch], c[0][ch]);
            const float c01 = fmaf(fw, c[3][ch] - c[2][ch], c[2][ch]);
            const float c10 = fmaf(fw, c[5][ch] - c[4][ch], c[4][ch]);
            const float c11 = fmaf(fw, c[7][ch] - c[6][ch], c[6][ch]);
            const float c0  = fmaf(fh, c01 - c00, c00);
            const float c1  = fmaf(fh, c11 - c10, c10);
            r[ch] = fmaf(fd, c1 - c0, c0);
        }

        pd = fmaf(scale, r[0], pd);
        ph = fmaf(scale, r[1], ph);
        pw = fmaf(scale, r[2], pw);
    }

    // results are written once and never re-read -> non-temporal stores
    __builtin_nontemporal_store(pd, out + 3 * t + 0);
    __builtin_nontemporal_store(ph, out + 3 * t + 1);
    __builtin_nontemporal_store(pw, out + 3 * t + 2);
}

// ---------------------------------------------------------------------------
// Host launcher
// ---------------------------------------------------------------------------
extern "C" void kernel_launch(void* const* d_in, const int* in_sizes, int n_in,
                              void* d_out, int out_size, void* d_ws, size_t ws_size,
                              hipStream_t stream) {
    const float* flow = (const float*)d_in[0];   // [2, 3, 128, 128, 128]
    const float* x    = (const float*)d_in[1];   // [2, 200000, 3]
    float*       out  = (float*)d_out;           // [2, 200000, 3]

    const int  npts_total = NBATCH * NPTS;
    const dim3 blk(256);
    const dim3 grd_int((npts_total + 255) / 256);
    const int  grd_pack = (NBATCH * NVOX) / 256;            // exact multiple

    const size_t need4 = (size_t)NBATCH * NVOX * sizeof(float4); // 64 MB
    const size_t need3 = (size_t)NBATCH * NVOX * 3 * sizeof(float); // 48 MB

    if (ws_size >= need4) {
        float4* vol = (float4*)d_ws;
        pack4_kernel<<<grd_pack, blk, 0, stream>>>(flow, vol);
        integrate_kernel<2><<<grd_int, blk, 0, stream>>>(nullptr, vol, x, out);
    } else if (ws_size >= need3) {
        float* vol = (float*)d_ws;
        pack3_kernel<<<grd_pack, blk, 0, stream>>>(flow, vol);
        integrate_kernel<1><<<grd_int, blk, 0, stream>>>(vol, nullptr, x, out);
    } else {
        integrate_kernel<0><<<grd_int, blk, 0, stream>>>(flow, nullptr, x, out);
    }
}